// CPAM_36429912604839
// MI455X (gfx1250) — compile-verified
//
#include <hip/hip_runtime.h>
#include <math.h>

typedef __attribute__((ext_vector_type(2))) float v2f;
typedef __attribute__((ext_vector_type(8))) float v8f;

// D = A(16x4, f32) x B(4x16, f32) + C(16x16 f32)
#define WMMA_F32(a, b, c) \
  __builtin_amdgcn_wmma_f32_16x16x4_f32(false, (a), false, (b), (short)0, (c), false, false)

// ---------------------------------------------------------------------------
// Generic implicit-GEMM conv (NCHW, OIHW), f32 WMMA. One wave per 16x16 tile.
// M = Cout (padded to 16), K = Cin*KH*KW (must be multiple of 4), N = OH*OW.
// ---------------------------------------------------------------------------
__global__ __launch_bounds__(32) void conv_wmma_kernel(
    const float* __restrict__ x, const float* __restrict__ w,
    const float* __restrict__ bias, float* __restrict__ out,
    int Cin, int Cout, int IH, int IW, int OH, int OW,
    int KH, int KW, int stride, int pad)
{
  const int lane = threadIdx.x;
  const int half = lane >> 4;       // 0: K lanes 0..1, 1: K lanes 2..3
  const int col  = lane & 15;
  const int b    = blockIdx.z;
  const int Mb   = blockIdx.y * 16;
  const int Nb   = blockIdx.x * 16;
  const int Np   = OH * OW;
  const int K    = Cin * KH * KW;
  const int KHW  = KH * KW;

  const int co_a = Mb + col;        // A-matrix row handled by this lane
  const int n_b  = Nb + col;        // B-matrix column handled by this lane
  const int oy   = n_b / OW;
  const int ox   = n_b - oy * OW;

  v8f acc = {};
  for (int k0 = 0; k0 < K; k0 += 4) {
    v2f a, bb;
#pragma unroll
    for (int r = 0; r < 2; ++r) {
      const int kk = k0 + 2 * half + r;
      a[r] = (co_a < Cout) ? w[co_a * K + kk] : 0.0f;
      const int ci  = kk / KHW;
      const int rem = kk - ci * KHW;
      const int ky  = rem / KW;
      const int kx  = rem - ky * KW;
      const int iy  = oy * stride - pad + ky;
      const int ix  = ox * stride - pad + kx;
      const bool ok = (iy >= 0) && (iy < IH) && (ix >= 0) && (ix < IW);
      bb[r] = ok ? x[(((size_t)b * Cin + ci) * IH + iy) * IW + ix] : 0.0f;
    }
    acc = WMMA_F32(a, bb, acc);
  }
#pragma unroll
  for (int r = 0; r < 8; ++r) {
    const int co = Mb + r + 8 * half;
    if (co < Cout)
      out[((size_t)b * Cout + co) * Np + n_b] = acc[r] + (bias ? bias[co] : 0.0f);
  }
}

// ---------------------------------------------------------------------------
// Training-mode BatchNorm: per-channel mean/var over (B,H,W) -> scale/shift
// ---------------------------------------------------------------------------
__global__ __launch_bounds__(256) void bn_stats_kernel(
    const float* __restrict__ x, const float* __restrict__ g,
    const float* __restrict__ beta, float* __restrict__ scale,
    float* __restrict__ shift, int Bn, int C, int HW)
{
  __shared__ float ssum[256], ssq[256];
  const int c = blockIdx.x, tid = threadIdx.x;
  float s = 0.f, q = 0.f;
  for (int b = 0; b < Bn; ++b) {
    const float* p = x + ((size_t)b * C + c) * HW;
    for (int i = tid; i < HW; i += 256) { const float v = p[i]; s += v; q += v * v; }
  }
  ssum[tid] = s; ssq[tid] = q; __syncthreads();
  for (int off = 128; off > 0; off >>= 1) {
    if (tid < off) { ssum[tid] += ssum[tid + off]; ssq[tid] += ssq[tid + off]; }
    __syncthreads();
  }
  if (tid == 0) {
    const float inv = 1.0f / (float)(Bn * HW);
    const float m   = ssum[0] * inv;
    const float var = ssq[0] * inv - m * m;
    const float sc  = g[c] * rsqrtf(var + 1e-5f);
    scale[c] = sc; shift[c] = beta[c] - m * sc;
  }
}

__global__ void bn_apply_kernel(const float* __restrict__ x,
                                const float* __restrict__ scale,
                                const float* __restrict__ shift,
                                float* __restrict__ y,
                                int total, int HW, int C, int relu)
{
  const int i = blockIdx.x * blockDim.x + threadIdx.x;
  if (i >= total) return;
  const int c = (i / HW) % C;
  float v = x[i] * scale[c] + shift[c];
  if (relu) v = fmaxf(v, 0.0f);
  y[i] = v;
}

// ---------------------------------------------------------------------------
// CAM: energy[c,d] = sum_n q[c,n]*k[d,n]   (M=N=64, K=1600)
// ---------------------------------------------------------------------------
__global__ __launch_bounds__(32) void cam_energy_kernel(
    const float* __restrict__ q, const float* __restrict__ k,
    float* __restrict__ e, int Np)
{
  const int lane = threadIdx.x, half = lane >> 4, col = lane & 15;
  const int b = blockIdx.z, Mb = blockIdx.y * 16, Nb = blockIdx.x * 16;
  const float* qb = q + (size_t)b * 64 * Np;
  const float* kb = k + (size_t)b * 64 * Np;
  v8f acc = {};
  for (int k0 = 0; k0 < Np; k0 += 4) {
    v2f a, bb;
#pragma unroll
    for (int r = 0; r < 2; ++r) {
      const int n = k0 + 2 * half + r;
      a[r]  = qb[(size_t)(Mb + col) * Np + n];
      bb[r] = kb[(size_t)(Nb + col) * Np + n];
    }
    acc = WMMA_F32(a, bb, acc);
  }
#pragma unroll
  for (int r = 0; r < 8; ++r)
    e[((size_t)b * 64 + Mb + r + 8 * half) * 64 + Nb + col] = acc[r];
}

// softmax(rowmax - e) == exp(rowmin - e)/sum  (one wave per 64-wide row)
__global__ __launch_bounds__(32) void cam_softmax_kernel(float* __restrict__ e)
{
  float* p = e + (size_t)blockIdx.x * 64;
  const int lane = threadIdx.x;
  const float e0 = p[lane], e1 = p[lane + 32];
  float rmin = fminf(e0, e1);
  for (int off = 1; off < 32; off <<= 1) rmin = fminf(rmin, __shfl_xor(rmin, off));
  const float p0 = __expf(rmin - e0), p1 = __expf(rmin - e1);
  float s = p0 + p1;
  for (int off = 1; off < 32; off <<= 1) s += __shfl_xor(s, off);
  p[lane] = p0 / s; p[lane + 32] = p1 / s;
}

// sa[c,n] = sum_d attn[c,d]*v[d,n]   (M=64, K=64, N=6400)
__global__ __launch_bounds__(32) void cam_apply_kernel(
    const float* __restrict__ attn, const float* __restrict__ v,
    float* __restrict__ sa, int Np)
{
  const int lane = threadIdx.x, half = lane >> 4, col = lane & 15;
  const int b = blockIdx.z, Mb = blockIdx.y * 16, Nb = blockIdx.x * 16;
  const float* ab = attn + (size_t)b * 64 * 64;
  const float* vb = v + (size_t)b * 64 * Np;
  v8f acc = {};
  for (int k0 = 0; k0 < 64; k0 += 4) {
    v2f a, bb;
#pragma unroll
    for (int r = 0; r < 2; ++r) {
      const int d = k0 + 2 * half + r;
      a[r]  = ab[(size_t)(Mb + col) * 64 + d];
      bb[r] = vb[(size_t)d * Np + Nb + col];
    }
    acc = WMMA_F32(a, bb, acc);
  }
#pragma unroll
  for (int r = 0; r < 8; ++r)
    sa[((size_t)b * 64 + Mb + r + 8 * half) * Np + Nb + col] = acc[r];
}

// ---------------------------------------------------------------------------
// PAM flash attention: sc[c,n] = sum_m softmax_m(pq^T pk)[n,m] * pv[c,m]
// 4 waves / block, each owns 16 query rows; K/V m-tiles staged in LDS once
// per block and shared by all 4 waves; online softmax; attn never stored.
// ---------------------------------------------------------------------------
__global__ __launch_bounds__(128) void pam_flash_kernel(
    const float* __restrict__ pq, const float* __restrict__ pk,
    const float* __restrict__ pv, float* __restrict__ sc, int Np)
{
  __shared__ float Vl[64 * 16];        // V tile,  Vl[c*16 + m]
  __shared__ float Kl[16 * 8];         // K tile,  Kl[m*8 + t]
  __shared__ float Pl[4][16 * 17];     // per-wave P transpose scratch

  const int tid  = threadIdx.x;
  const int wave = tid >> 5;
  const int lane = tid & 31;
  const int half = lane >> 4, col = lane & 15;
  const int b    = blockIdx.y;
  const int qb   = blockIdx.x * 64 + wave * 16;
  const float* pqb = pq + (size_t)b * 8 * Np;
  const float* pkb = pk + (size_t)b * 8 * Np;
  const float* pvb = pv + (size_t)b * 64 * Np;
  float* pls = Pl[wave];

  // staging roles (block-wide)
  const int vrow  = tid >> 1;          // 0..63 : channel c
  const int vhalf = (tid & 1) * 8;     // 8-float half of the 16-wide m row
  const int krow  = tid >> 4;          // 0..7  : t channel (tid < 128 -> 8x16)
  const int kcol  = tid & 15;          // m within tile

  v2f aq[2];                           // Q frag: A[M=qrow, K=t], K=8
#pragma unroll
  for (int j = 0; j < 2; ++j)
#pragma unroll
    for (int r = 0; r < 2; ++r) {
      const int t = 4 * j + 2 * half + r;
      aq[j][r] = pqb[(size_t)t * Np + qb + col];
    }

  float mrun[8], lrun[8];
#pragma unroll
  for (int r = 0; r < 8; ++r) { mrun[r] = -3.0e38f; lrun[r] = 0.0f; }
  v8f o[4] = {};                       // O[16 q-rows x 64 channels]

  for (int mb = 0; mb < Np; mb += 16) {
    // ---- cooperative staging of K/V tiles into LDS -----------------------
    const float* vsrc = pvb + (size_t)vrow * Np + mb + vhalf;
    const float4 v0 = *(const float4*)(vsrc + 0);
    const float4 v1 = *(const float4*)(vsrc + 4);
    *(float4*)&Vl[vrow * 16 + vhalf + 0] = v0;
    *(float4*)&Vl[vrow * 16 + vhalf + 4] = v1;
    if (krow < 8) Kl[kcol * 8 + krow] = pkb[(size_t)krow * Np + mb + kcol];
    // prefetch next m-tile of V into cache (global_prefetch_b8)
    if (mb + 16 < Np)
      __builtin_prefetch(pvb + (size_t)vrow * Np + mb + 16, 0, 1);
    __syncthreads();

    // ---- S = Q * K^T  (16x16, K=8 -> two x4 WMMAs) -----------------------
    v8f s = {};
#pragma unroll
    for (int j = 0; j < 2; ++j) {
      const v2f bk = *(const v2f*)&Kl[col * 8 + 4 * j + 2 * half];
      s = WMMA_F32(aq[j], bk, s);
    }

    // ---- online softmax: row reductions over the 16 lanes of each row ----
    float p[8];
#pragma unroll
    for (int r = 0; r < 8; ++r) {
      float rm = s[r];
      rm = fmaxf(rm, __shfl_xor(rm, 1));
      rm = fmaxf(rm, __shfl_xor(rm, 2));
      rm = fmaxf(rm, __shfl_xor(rm, 4));
      rm = fmaxf(rm, __shfl_xor(rm, 8));
      const float mnew  = fmaxf(mrun[r], rm);
      const float alpha = __expf(mrun[r] - mnew);
      const float pe    = __expf(s[r] - mnew);
      float rs = pe;
      rs += __shfl_xor(rs, 1); rs += __shfl_xor(rs, 2);
      rs += __shfl_xor(rs, 4); rs += __shfl_xor(rs, 8);
      lrun[r] = lrun[r] * alpha + rs;
      mrun[r] = mnew;
      p[r] = pe;
#pragma unroll
      for (int nt = 0; nt < 4; ++nt) o[nt][r] *= alpha;
    }

    // ---- C-layout -> A-layout transpose of P (wave-private LDS) ----------
#pragma unroll
    for (int r = 0; r < 8; ++r) pls[(r + 8 * half) * 17 + col] = p[r];
    v2f ap[4];
#pragma unroll
    for (int j = 0; j < 4; ++j)
#pragma unroll
      for (int r = 0; r < 2; ++r)
        ap[j][r] = pls[col * 17 + 4 * j + 2 * half + r];

    // ---- O += P * V ;  V[m, c] = Vl[c*16 + m] ----------------------------
#pragma unroll
    for (int nt = 0; nt < 4; ++nt) {
#pragma unroll
      for (int j = 0; j < 4; ++j) {
        const v2f bv = *(const v2f*)&Vl[(nt * 16 + col) * 16 + 4 * j + 2 * half];
        o[nt] = WMMA_F32(ap[j], bv, o[nt]);
      }
    }
    __syncthreads();   // protect Vl/Kl reads before next iteration's staging
  }
#pragma unroll
  for (int nt = 0; nt < 4; ++nt)
#pragma unroll
    for (int r = 0; r < 8; ++r)
      sc[((size_t)b * 64 + nt * 16 + col) * Np + qb + r + 8 * half] =
          o[nt][r] / lrun[r];
}

// feat_sum = sa*gamma1 + sc*gamma2 + feat1
__global__ void feat_sum_kernel(const float* __restrict__ sa,
                                const float* __restrict__ sc,
                                const float* __restrict__ f1,
                                const float* __restrict__ g1,
                                const float* __restrict__ g2,
                                float* __restrict__ out, int total)
{
  const int i = blockIdx.x * blockDim.x + threadIdx.x;
  if (i >= total) return;
  out[i] = sa[i] * g1[0] + sc[i] * g2[0] + f1[i];
}

// ---------------------------------------------------------------------------
extern "C" void kernel_launch(void* const* d_in, const int* in_sizes, int n_in,
                              void* d_out, int out_size, void* d_ws, size_t ws_size,
                              hipStream_t stream)
{
  (void)in_sizes; (void)n_in; (void)out_size; (void)ws_size;
  const float* x    = (const float*)d_in[0];
  const float* Wc   = (const float*)d_in[1];
  const float* bn1g = (const float*)d_in[2];
  const float* bn1b = (const float*)d_in[3];
  const float* cqw  = (const float*)d_in[4];
  const float* cqb  = (const float*)d_in[5];
  const float* ckw  = (const float*)d_in[6];
  const float* ckb  = (const float*)d_in[7];
  const float* cvw  = (const float*)d_in[8];
  const float* cvb  = (const float*)d_in[9];
  const float* pqw  = (const float*)d_in[10];
  const float* pqb  = (const float*)d_in[11];
  const float* pkw  = (const float*)d_in[12];
  const float* pkb  = (const float*)d_in[13];
  const float* pvw  = (const float*)d_in[14];
  const float* pvb  = (const float*)d_in[15];
  const float* g1   = (const float*)d_in[16];
  const float* g2   = (const float*)d_in[17];
  const float* bn2g = (const float*)d_in[18];
  const float* bn2b = (const float*)d_in[19];
  const float* Wd   = (const float*)d_in[20];
  const float* bn3g = (const float*)d_in[21];
  const float* bn3b = (const float*)d_in[22];

  const int Bn = 2, C = 64, H = 80, W = 80, HW = H * W, N = HW;  // N = 6400
  const int TOT = Bn * C * HW;                                   // 819200

  float* ws    = (float*)d_ws;
  float* t1    = ws;                     // conv_c out, later conv_d out
  float* feat1 = t1 + TOT;
  float* vbuf  = feat1 + TOT;            // CAM v, later feat_sum
  float* pvB   = vbuf + TOT;             // PAM v, later BN2 output
  float* saB   = pvB + TOT;
  float* scB   = saB + TOT;
  float* qB    = scB + TOT;              // [B,64,1600]
  float* kB    = qB + Bn * 64 * 1600;
  float* pqB   = kB + Bn * 64 * 1600;    // [B,8,6400]
  float* pkB   = pqB + Bn * 8 * N;
  float* eng   = pkB + Bn * 8 * N;       // [B,64,64]
  float* ss    = eng + Bn * 64 * 64;     // 3x (scale[64], shift[64])

  const dim3 blk32(32);
  // conv_c: 3x3 pad1 -> BN1 -> ReLU
  conv_wmma_kernel<<<dim3(N/16, C/16, Bn), blk32, 0, stream>>>(x, Wc, nullptr, t1, C, C, H, W, H, W, 3, 3, 1, 1);
  bn_stats_kernel<<<C, 256, 0, stream>>>(t1, bn1g, bn1b, ss + 0, ss + 64, Bn, C, HW);
  bn_apply_kernel<<<(TOT + 255)/256, 256, 0, stream>>>(t1, ss + 0, ss + 64, feat1, TOT, HW, C, 1);
  // CAM q,k (2x2 stride2), v (1x1)
  conv_wmma_kernel<<<dim3(1600/16, C/16, Bn), blk32, 0, stream>>>(feat1, cqw, cqb, qB, C, C, H, W, 40, 40, 2, 2, 2, 0);
  conv_wmma_kernel<<<dim3(1600/16, C/16, Bn), blk32, 0, stream>>>(feat1, ckw, ckb, kB, C, C, H, W, 40, 40, 2, 2, 2, 0);
  conv_wmma_kernel<<<dim3(N/16, C/16, Bn), blk32, 0, stream>>>(feat1, cvw, cvb, vbuf, C, C, H, W, H, W, 1, 1, 1, 0);
  // PAM pq, pk (C->8), pv (C->C), all 1x1
  conv_wmma_kernel<<<dim3(N/16, 1, Bn), blk32, 0, stream>>>(feat1, pqw, pqb, pqB, C, 8, H, W, H, W, 1, 1, 1, 0);
  conv_wmma_kernel<<<dim3(N/16, 1, Bn), blk32, 0, stream>>>(feat1, pkw, pkb, pkB, C, 8, H, W, H, W, 1, 1, 1, 0);
  conv_wmma_kernel<<<dim3(N/16, C/16, Bn), blk32, 0, stream>>>(feat1, pvw, pvb, pvB, C, C, H, W, H, W, 1, 1, 1, 0);
  // CAM attention
  cam_energy_kernel<<<dim3(4, 4, Bn), blk32, 0, stream>>>(qB, kB, eng, 1600);
  cam_softmax_kernel<<<Bn * C, blk32, 0, stream>>>(eng);
  cam_apply_kernel<<<dim3(N/16, C/16, Bn), blk32, 0, stream>>>(eng, vbuf, saB, N);
  // PAM flash attention (4 q-tiles per block, LDS-shared K/V tiles)
  pam_flash_kernel<<<dim3(N/64, Bn), dim3(128), 0, stream>>>(pqB, pkB, pvB, scB, N);
  // combine + BN2 -> conv_d -> BN3 -> ReLU
  feat_sum_kernel<<<(TOT + 255)/256, 256, 0, stream>>>(saB, scB, feat1, g1, g2, vbuf, TOT);
  bn_stats_kernel<<<C, 256, 0, stream>>>(vbuf, bn2g, bn2b, ss + 128, ss + 192, Bn, C, HW);
  bn_apply_kernel<<<(TOT + 255)/256, 256, 0, stream>>>(vbuf, ss + 128, ss + 192, pvB, TOT, HW, C, 0);
  conv_wmma_kernel<<<dim3(N/16, C/16, Bn), blk32, 0, stream>>>(pvB, Wd, nullptr, t1, C, C, H, W, H, W, 3, 3, 1, 1);
  bn_stats_kernel<<<C, 256, 0, stream>>>(t1, bn3g, bn3b, ss + 256, ss + 320, Bn, C, HW);
  bn_apply_kernel<<<(TOT + 255)/256, 256, 0, stream>>>(t1, ss + 256, ss + 320, (float*)d_out, TOT, HW, C, 1);
}